// Cvx_ShortestPathNet_1829656068355
// MI455X (gfx1250) — compile-verified
//
#include <hip/hip_runtime.h>
#include <hip/hip_bf16.h>

typedef __attribute__((ext_vector_type(2))) float v2f;
typedef __attribute__((ext_vector_type(8))) float v8f;

#define N1 400      // nodes
#define N2 760      // edges
#define BATCH 4096
#define CTX 512
#define HID 1024
#define NTOT (BATCH * N2)     // 3,112,960
#define RTOT (BATCH * N1)     // 1,638,400

// ---------------------------------------------------------------------------
// Utility: zero a float buffer
// ---------------------------------------------------------------------------
__global__ void zero_f32(float* __restrict__ p, int n) {
    int i = blockIdx.x * blockDim.x + threadIdx.x;
    if (i < n) p[i] = 0.0f;
}

// ---------------------------------------------------------------------------
// Build K = A*A^T + (1/n) * ones  (400x400 SPD).  inv(K) acts as pinv(L) on
// vectors orthogonal to ones (always true for our RHS).
// ---------------------------------------------------------------------------
__global__ void build_K(const float* __restrict__ A, float* __restrict__ K) {
    int idx = blockIdx.x * blockDim.x + threadIdx.x;
    if (idx >= N1 * N1) return;
    int i = idx / N1, j = idx % N1;
    float s = 0.0f;
    const float* ai = A + (size_t)i * N2;
    const float* aj = A + (size_t)j * N2;
    for (int e = 0; e < N2; ++e) s += ai[e] * aj[e];
    K[idx] = s + (1.0f / (float)N1);
}

// ---------------------------------------------------------------------------
// In-place Gauss-Jordan sweep, pivot p (no pivoting needed: K is SPD).
// Phase 1 (single block): save column p, scale row p, store d = 1/K[p][p].
// ---------------------------------------------------------------------------
__global__ void gj_prep(float* __restrict__ K, float* __restrict__ colbuf, int p) {
    int t = threadIdx.x;
    for (int i = t; i < N1; i += blockDim.x) colbuf[i] = K[(size_t)i * N1 + p];
    __syncthreads();
    float d = 1.0f / colbuf[p];
    if (t == 0) colbuf[N1] = d;
    for (int j = t; j < N1; j += blockDim.x)
        if (j != p) K[(size_t)p * N1 + j] *= d;
}

// Phase 2 (grid): rank-1 update of interior, rewrite column p, set diagonal.
__global__ void gj_update(float* __restrict__ K, const float* __restrict__ colbuf, int p) {
    int idx = blockIdx.x * blockDim.x + threadIdx.x;
    if (idx >= N1 * N1) return;
    int i = idx / N1, j = idx % N1;
    float d = colbuf[N1];
    if (i == p) {
        if (j == p) K[idx] = d;          // row p (j!=p) already scaled in phase 1
    } else if (j == p) {
        K[idx] = -colbuf[i] * d;
    } else {
        K[idx] -= colbuf[i] * K[(size_t)p * N1 + j];
    }
}

// ---------------------------------------------------------------------------
// Sparse structure extraction from incidence matrix A [400 x 760].
// Each edge column has exactly one +1 (tail) and one -1 (head).
// ---------------------------------------------------------------------------
__global__ void edge_setup(const float* __restrict__ A,
                           int* __restrict__ eu, int* __restrict__ ev) {
    int e = blockIdx.x * blockDim.x + threadIdx.x;
    if (e >= N2) return;
    int u = 0, v = 0;
    for (int i = 0; i < N1; ++i) {
        float a = A[(size_t)i * N2 + e];
        if (a > 0.5f) u = i;
        else if (a < -0.5f) v = i;
    }
    eu[e] = u; ev[e] = v;
}

__global__ void node_setup(const float* __restrict__ A,
                           int* __restrict__ nidx, float* __restrict__ nsgn) {
    int i = blockIdx.x * blockDim.x + threadIdx.x;
    if (i >= N1) return;
    int cnt = 0;
    for (int e = 0; e < N2 && cnt < 4; ++e) {
        float a = A[(size_t)i * N2 + e];
        if (a > 0.5f || a < -0.5f) {
            nidx[i * 4 + cnt] = e;
            nsgn[i * 4 + cnt] = a;
            ++cnt;
        }
    }
    for (; cnt < 4; ++cnt) { nidx[i * 4 + cnt] = -1; nsgn[i * 4 + cnt] = 0.0f; }
}

// ---------------------------------------------------------------------------
// WMMA f32 GEMM:  C[M,N] = A[M,K] @ B[N,K]^T   (all row-major)
// MODE 0: plain   MODE 1: +bias, LeakyReLU(0.1)   MODE 2: +bias
// Block = 128 threads (4 wave32), computes a 128(M) x 64(N) tile.
// Wave w owns rows [w*32, w*32+32): 2 A-fragments x 4 N-subtiles
// -> 8 accumulators, 0.75 LDS fragment loads per WMMA.
// A/B staged via padded LDS; next K-tile prefetched during compute.
// Requires: M % 128 == 0 (true here: M = 4096 everywhere).
// ---------------------------------------------------------------------------
template <int MODE>
__global__ __launch_bounds__(128)
void wmma_gemm_nt(const float* __restrict__ Amat, const float* __restrict__ Bmat,
                  const float* __restrict__ bias, float* __restrict__ Cmat,
                  int Mdim, int Ndim, int Kdim) {
    __shared__ float As[128][33];
    __shared__ float Bs[64][33];

    const int tid  = threadIdx.x;
    const int wave = tid >> 5;
    const int lane = tid & 31;
    const int mblk = blockIdx.y * 128;
    const int nblk = blockIdx.x * 64;

    v8f acc[2][4];
#pragma unroll
    for (int s = 0; s < 2; ++s)
#pragma unroll
        for (int t = 0; t < 4; ++t)
#pragma unroll
            for (int q = 0; q < 8; ++q) acc[s][t][q] = 0.0f;

    const int m   = lane & 15;
    const int klo = (lane < 16) ? 0 : 2;

    for (int k0 = 0; k0 < Kdim; k0 += 32) {
        // cooperative stage: A 128x32, B 64x32 (B used transposed)
#pragma unroll
        for (int pass = 0; pass < 32; ++pass) {
            int idx = tid + pass * 128;        // 0..4095
            int r = idx >> 5, c = idx & 31;
            int gk = k0 + c;
            int gm = mblk + r;
            As[r][c] = (gk < Kdim && gm < Mdim) ? Amat[(size_t)gm * Kdim + gk] : 0.0f;
        }
#pragma unroll
        for (int pass = 0; pass < 16; ++pass) {
            int idx = tid + pass * 128;        // 0..2047
            int r = idx >> 5, c = idx & 31;
            int gk = k0 + c;
            int gn = nblk + r;
            Bs[r][c] = (gk < Kdim && gn < Ndim) ? Bmat[(size_t)gn * Kdim + gk] : 0.0f;
        }
        // prefetch next K-tile (one line per row) -> global_prefetch_b8
        if (k0 + 32 < Kdim) {
            int rA = mblk + tid;
            if (rA < Mdim)
                __builtin_prefetch(Amat + (size_t)rA * Kdim + k0 + 32, 0, 1);
            int rB = nblk + (tid & 63);
            if (tid < 64 && rB < Ndim)
                __builtin_prefetch(Bmat + (size_t)rB * Kdim + k0 + 32, 0, 1);
        }
        __syncthreads();

#pragma unroll
        for (int kk = 0; kk < 32; kk += 4) {
            v2f a0, a1;
            a0.x = As[wave * 32 + m][kk + klo];
            a0.y = As[wave * 32 + m][kk + klo + 1];
            a1.x = As[wave * 32 + 16 + m][kk + klo];
            a1.y = As[wave * 32 + 16 + m][kk + klo + 1];
#pragma unroll
            for (int t = 0; t < 4; ++t) {
                v2f b;
                b.x = Bs[t * 16 + m][kk + klo];
                b.y = Bs[t * 16 + m][kk + klo + 1];
                acc[0][t] = __builtin_amdgcn_wmma_f32_16x16x4_f32(
                    false, a0, false, b, (short)0, acc[0][t], false, false);
                acc[1][t] = __builtin_amdgcn_wmma_f32_16x16x4_f32(
                    false, a1, false, b, (short)0, acc[1][t], false, false);
            }
        }
        __syncthreads();
    }

    // epilogue: C/D layout: VGPR q -> row q (lanes 0-15) / q+8 (lanes 16-31)
    const int rhi = (lane < 16) ? 0 : 8;
#pragma unroll
    for (int s = 0; s < 2; ++s) {
        const int mrow = mblk + wave * 32 + s * 16;
#pragma unroll
        for (int t = 0; t < 4; ++t) {
            int gn = nblk + t * 16 + (lane & 15);
            if (gn >= Ndim) continue;
            float bv = (MODE != 0) ? bias[gn] : 0.0f;
#pragma unroll
            for (int q = 0; q < 8; ++q) {
                int gm = mrow + q + rhi;
                float val = acc[s][t][q] + bv;
                if (MODE == 1) val = (val >= 0.0f) ? val : 0.1f * val;
                if (gm < Mdim) Cmat[(size_t)gm * Ndim + gn] = val;
            }
        }
    }
}

// ---------------------------------------------------------------------------
// ADMM elementwise / sparse kernels (RHO = 1)
// ---------------------------------------------------------------------------
// v = (z - u) - w
__global__ void compute_v(const float* __restrict__ w, const float* __restrict__ z,
                          const float* __restrict__ u, float* __restrict__ v) {
    int i = blockIdx.x * blockDim.x + threadIdx.x;
    if (i < NTOT) v[i] = (z[i] - u[i]) - w[i];
}

// r[b,i] = sum_{edges e incident to i} sign(i,e) * v[b,e] - 2*b_eq[i]
__global__ void compute_r(const float* __restrict__ v, const float* __restrict__ beq,
                          const int* __restrict__ nidx, const float* __restrict__ nsgn,
                          float* __restrict__ r) {
    int idx = blockIdx.x * blockDim.x + threadIdx.x;
    if (idx >= RTOT) return;
    int i = idx % N1;
    int b = idx / N1;
    const float* vb = v + (size_t)b * N2;
    float acc = -2.0f * beq[i];
#pragma unroll
    for (int s = 0; s < 4; ++s) {
        int e = nidx[i * 4 + s];
        if (e >= 0) acc += nsgn[i * 4 + s] * vb[e];
    }
    r[idx] = acc;
}

// x = (v - (nu[eu] - nu[ev])) * 0.5 ; z = max(x+u,0) ; u += x - z ; x -> out
__global__ void admm_update(const float* __restrict__ v, const float* __restrict__ nu,
                            const int* __restrict__ eu, const int* __restrict__ ev,
                            float* __restrict__ z, float* __restrict__ u,
                            float* __restrict__ xout) {
    int idx = blockIdx.x * blockDim.x + threadIdx.x;
    if (idx >= NTOT) return;
    int e = idx % N2;
    int b = idx / N2;
    const float* nub = nu + (size_t)b * N1;
    float natae = nub[eu[e]] - nub[ev[e]];
    float x  = 0.5f * (v[idx] - natae);
    float uu = u[idx];
    float zn = x + uu; zn = (zn > 0.0f) ? zn : 0.0f;
    z[idx] = zn;
    u[idx] = uu + x - zn;
    xout[idx] = x;
}

// ---------------------------------------------------------------------------
// Host orchestration
// ---------------------------------------------------------------------------
extern "C" void kernel_launch(void* const* d_in, const int* in_sizes, int n_in,
                              void* d_out, int out_size, void* d_ws, size_t ws_size,
                              hipStream_t stream) {
    const float* d_d  = (const float*)d_in[0];   // [4096,512]
    const float* d_W1 = (const float*)d_in[1];   // [1024,512]
    const float* d_b1 = (const float*)d_in[2];   // [1024]
    const float* d_W2 = (const float*)d_in[3];   // [760,1024]
    const float* d_b2 = (const float*)d_in[4];   // [760]
    const float* d_A  = (const float*)d_in[5];   // [400,760]
    const float* d_be = (const float*)d_in[6];   // [400]
    float* xout = (float*)d_out;                 // [4096,760]

    // workspace carve-up (floats)
    float* ws     = (float*)d_ws;
    float* Kmat   = ws;                          // 160,000
    float* colbuf = Kmat + 160000;               // 512
    int*   eu     = (int*)(colbuf + 512);        // 768
    int*   ev     = eu + 768;                    // 768
    int*   nidx   = ev + 768;                    // 1,600
    float* nsgn   = (float*)(nidx + 1600);       // 1,600
    float* Hbuf   = nsgn + 1664;                 // 4,194,304 (reused as v later)
    float* wbuf   = Hbuf + 4194304;              // 3,112,960
    float* zbuf   = wbuf + NTOT;                 // 3,112,960
    float* ubuf   = zbuf + NTOT;                 // 3,112,960
    float* rbuf   = ubuf + NTOT;                 // 1,638,400
    float* nubuf  = rbuf + RTOT;                 // 1,638,400
    float* vbuf   = Hbuf;                        // H dead after GEMM2

    // ---- one-time setup: K = A A^T + J/n, Gauss-Jordan inverse, sparsity ----
    build_K<<<(N1 * N1 + 255) / 256, 256, 0, stream>>>(d_A, Kmat);
    for (int p = 0; p < N1; ++p) {
        gj_prep  <<<1, 512, 0, stream>>>(Kmat, colbuf, p);
        gj_update<<<(N1 * N1 + 255) / 256, 256, 0, stream>>>(Kmat, colbuf, p);
    }
    edge_setup<<<(N2 + 255) / 256, 256, 0, stream>>>(d_A, eu, ev);
    node_setup<<<(N1 + 255) / 256, 256, 0, stream>>>(d_A, nidx, nsgn);

    // ---- MLP router (fp32 WMMA) ----
    wmma_gemm_nt<1><<<dim3(HID / 64, BATCH / 128), 128, 0, stream>>>(
        d_d, d_W1, d_b1, Hbuf, BATCH, HID, CTX);
    wmma_gemm_nt<2><<<dim3((N2 + 63) / 64, BATCH / 128), 128, 0, stream>>>(
        Hbuf, d_W2, d_b2, wbuf, BATCH, N2, HID);

    // ---- ADMM state init ----
    zero_f32<<<(NTOT + 255) / 256, 256, 0, stream>>>(zbuf, NTOT);
    zero_f32<<<(NTOT + 255) / 256, 256, 0, stream>>>(ubuf, NTOT);

    // ---- 50 ADMM iterations ----
    for (int it = 0; it < 50; ++it) {
        compute_v<<<(NTOT + 255) / 256, 256, 0, stream>>>(wbuf, zbuf, ubuf, vbuf);
        compute_r<<<(RTOT + 255) / 256, 256, 0, stream>>>(vbuf, d_be, nidx, nsgn, rbuf);
        // nu = r @ inv(K)  (inv(K) symmetric -> NT GEMM is exact)
        wmma_gemm_nt<0><<<dim3((N1 + 63) / 64, BATCH / 128), 128, 0, stream>>>(
            rbuf, Kmat, nullptr, nubuf, BATCH, N1, N1);
        admm_update<<<(NTOT + 255) / 256, 256, 0, stream>>>(
            vbuf, nubuf, eu, ev, zbuf, ubuf, xout);
    }
}